// Head_60086592471088
// MI455X (gfx1250) — compile-verified
//
#include <hip/hip_runtime.h>
#include <hip/hip_bf16.h>

// ---------------------------------------------------------------------------
// Problem constants (from reference): N=4096 E=8192 L=65536 DN=256 DE=128
// ---------------------------------------------------------------------------
#define NN 4096
#define EE 8192
#define LL 65536
#define DNN 256
#define DEE 128

typedef __attribute__((ext_vector_type(2))) float v2f;
typedef __attribute__((ext_vector_type(4))) float v4f;
typedef __attribute__((ext_vector_type(8))) float v8f;

// ---------------------------------------------------------------------------
// out[i] = dot(x[i,:], w) + *bias      (one wave32 per row, 8 rows per block)
// ---------------------------------------------------------------------------
__global__ __launch_bounds__(256) void rowdot_kernel(
    const float* __restrict__ x, const float* __restrict__ w,
    const float* __restrict__ bias, int d, float* __restrict__ out) {
  int lane = threadIdx.x & 31;
  int wid  = threadIdx.x >> 5;
  int row  = blockIdx.x * 8 + wid;
  const float* xr = x + (size_t)row * d;
  float acc = 0.f;
  for (int j = lane; j < d; j += 32) acc += xr[j] * w[j];
#pragma unroll
  for (int off = 16; off > 0; off >>= 1) acc += __shfl_xor(acc, off, 32);
  if (lane == 0) out[row] = acc + bias[0];
}

// ---------------------------------------------------------------------------
// t[c] += sum over a 64-row slab of x[r,c]*k[r]      (blockDim.x == d)
// ---------------------------------------------------------------------------
__global__ void colsum_kernel(const float* __restrict__ x,
                              const float* __restrict__ k, int d,
                              float* __restrict__ t) {
  int col = threadIdx.x;
  int r0  = blockIdx.x * 64;
  float acc = 0.f;
#pragma unroll 4
  for (int r = 0; r < 64; ++r)
    acc += x[(size_t)(r0 + r) * d + col] * k[r0 + r];
  atomicAdd(&t[col], acc);
}

// ---------------------------------------------------------------------------
// u = vw @ t (row dots), c_out = dot(vb, t)          (one block, blockDim == d)
// ---------------------------------------------------------------------------
__global__ void smallmv_kernel(const float* __restrict__ vw,
                               const float* __restrict__ vb,
                               const float* __restrict__ t, int d,
                               float* __restrict__ u, float* __restrict__ c_out) {
  __shared__ float red[256];
  int tid = threadIdx.x;
  float acc = 0.f;
  for (int j = 0; j < d; ++j) acc += vw[(size_t)tid * d + j] * t[j];
  u[tid] = acc;
  red[tid] = vb[tid] * t[tid];
  __syncthreads();
  for (int off = d >> 1; off > 0; off >>= 1) {
    if (tid < off) red[tid] += red[tid + off];
    __syncthreads();
  }
  if (tid == 0) *c_out = red[0];
}

// ---------------------------------------------------------------------------
// att = softmax(s) over n, single 1024-thread block
// ---------------------------------------------------------------------------
__global__ __launch_bounds__(1024) void softmax_kernel(
    const float* __restrict__ s, int n, float* __restrict__ att) {
  __shared__ float red[1024];
  __shared__ float sMax, sInv;
  int tid = threadIdx.x;
  float m = -3.4e38f;
  for (int i = tid; i < n; i += 1024) m = fmaxf(m, s[i]);
  red[tid] = m;
  __syncthreads();
  for (int off = 512; off > 0; off >>= 1) {
    if (tid < off) red[tid] = fmaxf(red[tid], red[tid + off]);
    __syncthreads();
  }
  if (tid == 0) sMax = red[0];
  __syncthreads();
  float acc = 0.f;
  for (int i = tid; i < n; i += 1024) acc += __expf(s[i] - sMax);
  red[tid] = acc;
  __syncthreads();
  for (int off = 512; off > 0; off >>= 1) {
    if (tid < off) red[tid] += red[tid + off];
    __syncthreads();
  }
  if (tid == 0) sInv = 1.0f / red[0];
  __syncthreads();
  for (int i = tid; i < n; i += 1024) att[i] = __expf(s[i] - sMax) * sInv;
}

// ---------------------------------------------------------------------------
// deg[src[e]] += 1
// ---------------------------------------------------------------------------
__global__ void deg_kernel(const int* __restrict__ src, int e,
                           float* __restrict__ deg) {
  int i = blockIdx.x * blockDim.x + threadIdx.x;
  if (i < e) atomicAdd(&deg[src[i]], 1.0f);
}

// ---------------------------------------------------------------------------
// rdeg[i] = 1 / deg[i]   (deg >= 1 guaranteed by the reference construction)
// ---------------------------------------------------------------------------
__global__ void recip_kernel(const float* __restrict__ deg, int n,
                             float* __restrict__ rdeg) {
  int i = blockIdx.x * blockDim.x + threadIdx.x;
  if (i < n) rdeg[i] = __frcp_rn(deg[i]);
}

// ---------------------------------------------------------------------------
// agg[src[e], :] += att[e] * q[dst[e], :]     (DN = 256 columns)
// ---------------------------------------------------------------------------
__global__ __launch_bounds__(256) void scatter_node_kernel(
    const int* __restrict__ src, const int* __restrict__ dst,
    const float* __restrict__ att, const float* __restrict__ q,
    float* __restrict__ agg) {
  int idx = blockIdx.x * 256 + threadIdx.x;   // E*DN = 2^21 threads
  int e = idx >> 8;
  int c = idx & 255;
  float w = att[e];
  int d = dst[e];
  atomicAdd(&agg[(size_t)src[e] * DNN + c], w * q[(size_t)d * DNN + c]);
}

// ---------------------------------------------------------------------------
// agg2[lg_src[l], :] += node_att[dst[lg_src[l]]] * q[lg_dst[l], :]  (DE = 128)
// ---------------------------------------------------------------------------
__global__ __launch_bounds__(256) void scatter_edge_kernel(
    const int* __restrict__ lg_src, const int* __restrict__ lg_dst,
    const int* __restrict__ dst, const float* __restrict__ node_att,
    const float* __restrict__ q, float* __restrict__ agg2) {
  int idx = blockIdx.x * 256 + threadIdx.x;   // L*DE = 2^23 threads
  int l = idx >> 7;
  int c = idx & 127;
  int se = lg_src[l];
  float w = node_att[dst[se]];
  atomicAdd(&agg2[(size_t)se * DEE + c], w * q[(size_t)lg_dst[l] * DEE + c]);
}

// ---------------------------------------------------------------------------
// C = diag(rscale) * A @ W + bias   via V_WMMA_F32_16X16X4_F32
//   block: 256 threads = 8 waves; block tile 64(M) x 32(N); K chunked by 32
//   wave (wm in 0..3, wn in 0..1) owns one 16x16 C tile, v8f accumulator.
//   A-frag per ISA 7.12.2: lane L -> M = L%16 ; v0/v1 hold K = {0,1}+2*(L>=16)
//   Staging is b128: A thread covers rows (tid>>3, tid>>3+32), cols (tid&7)*4;
//   the per-row scale is loop-invariant and hoisted out of the K loop.
// ---------------------------------------------------------------------------
template <bool SCALED>
__global__ __launch_bounds__(256) void gemm_wmma_kernel(
    const float* __restrict__ A, const float* __restrict__ rscale,
    const float* __restrict__ W, const float* __restrict__ bias,
    float* __restrict__ C, int M, int N, int K) {
  __shared__ float As[64 * 36];   // stride 36 floats = 144B: 16B-aligned rows
  __shared__ float Bs[32 * 36];

  const int tid  = threadIdx.x;
  const int lane = tid & 31;
  const int wid  = tid >> 5;
  const int wm   = wid & 3;        // 0..3 -> M sub-tile
  const int wn   = wid >> 2;       // 0..1 -> N sub-tile
  const int half = lane >> 4;      // 0: lanes 0-15, 1: lanes 16-31
  const int mrow = lane & 15;
  const int gM = blockIdx.x * 64;
  const int gN = blockIdx.y * 32;

  // staging geometry (same rows for every K chunk)
  const int rS = tid >> 3;         // 0..31
  const int cS = (tid & 7) * 4;    // 0,4,..,28
  float s0 = 1.f, s1 = 1.f;
  if (SCALED) {
    s0 = rscale[gM + rS];
    s1 = rscale[gM + rS + 32];
  }
  const float* Arow0 = A + (size_t)(gM + rS) * K;
  const float* Arow1 = A + (size_t)(gM + rS + 32) * K;

  v8f acc = {0.f, 0.f, 0.f, 0.f, 0.f, 0.f, 0.f, 0.f};

  for (int k0 = 0; k0 < K; k0 += 32) {
    // ---- stage A 64x32: two b128 per thread ----
    v4f a0 = *(const v4f*)(Arow0 + k0 + cS);
    v4f a1 = *(const v4f*)(Arow1 + k0 + cS);
    if (SCALED) {
      a0.x *= s0; a0.y *= s0; a0.z *= s0; a0.w *= s0;
      a1.x *= s1; a1.y *= s1; a1.z *= s1; a1.w *= s1;
    }
    *(v4f*)&As[rS * 36 + cS] = a0;
    *(v4f*)&As[(rS + 32) * 36 + cS] = a1;
    // ---- stage B 32x32: one b128 per thread ----
    *(v4f*)&Bs[rS * 36 + cS] =
        *(const v4f*)&W[(size_t)(k0 + rS) * N + (gN + cS)];
    if (k0 + 32 < K) {  // speculative prefetch of next A/B chunks
      __builtin_prefetch(Arow0 + k0 + 32 + cS, 0, 0);
      __builtin_prefetch(&W[(size_t)(k0 + 32 + rS) * N + gN + cS], 0, 0);
    }
    __syncthreads();

#pragma unroll
    for (int kk = 0; kk < 32; kk += 4) {
      const int ko = kk + 2 * half;
      v2f a, b;
      const int am = wm * 16 + mrow;
      a.x = As[am * 36 + ko];
      a.y = As[am * 36 + ko + 1];
      const int bn = wn * 16 + mrow;
      b.x = Bs[ko * 36 + bn];
      b.y = Bs[(ko + 1) * 36 + bn];
      acc = __builtin_amdgcn_wmma_f32_16x16x4_f32(
          /*neg_a=*/false, a, /*neg_b=*/false, b,
          /*c_mod=*/(short)0, acc, /*reuse_a=*/false, /*reuse_b=*/false);
    }
    __syncthreads();
  }

  // ---- epilogue: VGPR r holds (M = r + 8*half, N = mrow) of the 16x16 tile
  const int col = gN + wn * 16 + mrow;
  const float bv = bias[col];
#pragma unroll
  for (int r = 0; r < 8; ++r) {
    int row = gM + wm * 16 + r + half * 8;
    C[(size_t)row * N + col] = acc[r] + bv;
  }
}

// ---------------------------------------------------------------------------
// Host-side orchestration
// ---------------------------------------------------------------------------
extern "C" void kernel_launch(void* const* d_in, const int* in_sizes, int n_in,
                              void* d_out, int out_size, void* d_ws, size_t ws_size,
                              hipStream_t stream) {
  const float* node_feats = (const float*)d_in[0];
  const float* edge_feats = (const float*)d_in[1];
  const int*   src    = (const int*)d_in[2];
  const int*   dst    = (const int*)d_in[3];
  const int*   lg_src = (const int*)d_in[4];
  const int*   lg_dst = (const int*)d_in[5];
  const float* nk_w = (const float*)d_in[6];
  const float* nk_b = (const float*)d_in[7];
  const float* nv_w = (const float*)d_in[8];
  const float* nv_b = (const float*)d_in[9];
  const float* nq_w = (const float*)d_in[10];
  const float* nq_b = (const float*)d_in[11];
  const float* nc_w = (const float*)d_in[12];
  const float* nc_b = (const float*)d_in[13];
  const float* ek_w = (const float*)d_in[14];
  const float* ek_b = (const float*)d_in[15];
  const float* ev_w = (const float*)d_in[16];
  const float* ev_b = (const float*)d_in[17];
  const float* eq_w = (const float*)d_in[18];
  const float* eq_b = (const float*)d_in[19];
  const float* ec_w = (const float*)d_in[20];
  const float* ec_b = (const float*)d_in[21];

  float* ws = (float*)d_ws;
  // workspace layout (float offsets)
  float* k_n   = ws + 0;            // N
  float* k_e   = ws + 4096;         // E
  float* s_n   = ws + 12288;        // N
  float* s_e   = ws + 16384;        // E
  float* att_n = ws + 24576;        // N
  float* att_e = ws + 28672;        // E
  float* t_n   = ws + 36864;        // DN
  float* t_e   = ws + 37120;        // DE
  float* u_n   = ws + 37248;        // DN
  float* u_e   = ws + 37504;        // DE
  float* c_n   = ws + 37632;        // 1
  float* c_e   = ws + 37633;        // 1
  float* deg   = ws + 37640;        // N
  float* rdeg  = ws + 41736;        // N
  float* q1_n  = ws + 65536;        // N*DN  = 1048576
  float* agg_n = ws + 65536 + 1048576;      // N*DN
  float* q1_e  = ws + 65536 + 2 * 1048576;  // E*DE  = 1048576
  float* agg_e = ws + 65536 + 3 * 1048576;  // E*DE

  float* node_out = (float*)d_out;                      // N*DN
  float* edge_out = (float*)d_out + (size_t)NN * DNN;   // E*DE

  // zero the atomic-add targets
  hipMemsetAsync(t_n, 0, DNN * sizeof(float), stream);
  hipMemsetAsync(t_e, 0, DEE * sizeof(float), stream);
  hipMemsetAsync(deg, 0, NN * sizeof(float), stream);
  hipMemsetAsync(agg_n, 0, (size_t)NN * DNN * sizeof(float), stream);
  hipMemsetAsync(agg_e, 0, (size_t)EE * DEE * sizeof(float), stream);

  // ---- attention, algebraically reduced to matvecs ----
  rowdot_kernel<<<NN / 8, 256, 0, stream>>>(node_feats, nk_w, nk_b, DNN, k_n);
  rowdot_kernel<<<EE / 8, 256, 0, stream>>>(edge_feats, ek_w, ek_b, DEE, k_e);
  colsum_kernel<<<NN / 64, DNN, 0, stream>>>(node_feats, k_n, DNN, t_n);
  colsum_kernel<<<EE / 64, DEE, 0, stream>>>(edge_feats, k_e, DEE, t_e);
  smallmv_kernel<<<1, DNN, 0, stream>>>(nv_w, nv_b, t_n, DNN, u_n, c_n);
  smallmv_kernel<<<1, DEE, 0, stream>>>(ev_w, ev_b, t_e, DEE, u_e, c_e);
  rowdot_kernel<<<NN / 8, 256, 0, stream>>>(node_feats, u_n, c_n, DNN, s_n);
  rowdot_kernel<<<EE / 8, 256, 0, stream>>>(edge_feats, u_e, c_e, DEE, s_e);
  softmax_kernel<<<1, 1024, 0, stream>>>(s_n, NN, att_n);
  softmax_kernel<<<1, 1024, 0, stream>>>(s_e, EE, att_e);

  // ---- query GEMMs (WMMA fp32) ----
  gemm_wmma_kernel<false><<<dim3(NN / 64, DNN / 32), 256, 0, stream>>>(
      node_feats, nullptr, nq_w, nq_b, q1_n, NN, DNN, DNN);
  gemm_wmma_kernel<false><<<dim3(EE / 64, DEE / 32), 256, 0, stream>>>(
      edge_feats, nullptr, eq_w, eq_b, q1_e, EE, DEE, DEE);

  // ---- sparse aggregations ----
  deg_kernel<<<EE / 256, 256, 0, stream>>>(src, EE, deg);
  recip_kernel<<<NN / 256, 256, 0, stream>>>(deg, NN, rdeg);
  scatter_node_kernel<<<(EE * DNN) / 256, 256, 0, stream>>>(
      src, dst, att_e, q1_n, agg_n);
  scatter_edge_kernel<<<(LL / 256) * DEE, 256, 0, stream>>>(
      lg_src, lg_dst, dst, att_n, q1_e, agg_e);

  // ---- conv GEMMs (WMMA fp32); node path folds the /deg row-scale ----
  gemm_wmma_kernel<true><<<dim3(NN / 64, DNN / 32), 256, 0, stream>>>(
      agg_n, rdeg, nc_w, nc_b, node_out, NN, DNN, DNN);
  gemm_wmma_kernel<false><<<dim3(EE / 64, DEE / 32), 256, 0, stream>>>(
      agg_e, nullptr, ec_w, ec_b, edge_out, EE, DEE, DEE);
}